// Diffusion_38929583571393
// MI455X (gfx1250) — compile-verified
//
#include <hip/hip_runtime.h>

#define ZDIM 64
#define YDIM 1024
#define XDIM 1024
#define COEFF 0.05f   // ALPHA * DT
#define ITERS 10

typedef __attribute__((ext_vector_type(2))) float v2f;
typedef __attribute__((ext_vector_type(8))) float v8f;
typedef int v4i __attribute__((vector_size(16)));
typedef __attribute__((address_space(3))) v4i lds_v4i;

#ifndef __has_builtin
#define __has_builtin(x) 0
#endif

#if __has_builtin(__builtin_amdgcn_global_load_async_to_lds_b128)
#define USE_ASYNC_LDS 1
#endif

#if __has_builtin(__builtin_amdgcn_s_wait_asynccnt)
#define WAIT_ASYNC() __builtin_amdgcn_s_wait_asynccnt(0)
#else
#define WAIT_ASYNC() asm volatile("s_wait_asynccnt 0" ::: "memory")
#endif

__device__ __forceinline__ int clampi(int v, int lo, int hi) {
    return v < lo ? lo : (v > hi ? hi : v);
}

// Exact central-difference operator with halo: output row i of a 16-row tile
// reads halo-extended rows k (k = y_local + 1, y_local in -1..16):
//   dy[i] = 0.5*(X[i+1] - X[i-1]) = 0.5*B[k=i+2] - 0.5*B[k=i]
__device__ __forceinline__ float gcoef(int i, int k) {
    return (k == i) ? -0.5f : ((k == i + 2) ? 0.5f : 0.0f);
}

__global__ __launch_bounds__(128) void diffusion_step(const float* __restrict__ src,
                                                      float* __restrict__ dst) {
    // 16 rows x (1 left halo + 64 center + 1 right halo), stride 76 floats:
    //   [3] = left halo, [4..67] = center (16B aligned), [68] = right halo.
    // stride 76 => the two WMMA lane-halves (rows r and r+8) land on
    // disjoint LDS bank groups (8*76 mod 64 == 32).
    __shared__ __align__(16) float smem[16 * 76];

    const int t    = threadIdx.x;
    const int lane = t & 31;
    const int w    = t >> 5;           // wave id 0..3
    const int hi   = (lane >> 4) & 1;  // lane half within wave
    const int col  = lane & 15;        // column (and A-matrix row index)

    const int xb0 = blockIdx.x * 64;   // block x base
    const int yb  = blockIdx.y * 16;   // tile y base
    const int z   = blockIdx.z;

    const size_t plane = (size_t)YDIM * XDIM;
    const size_t zoff  = (size_t)z * plane;

    // ---- stage center 16x64 tile into LDS (async global->LDS if available) ----
    // 256 float4 chunks, each thread moves q = t and q = t + 128.
#pragma unroll
    for (int s = 0; s < 2; ++s) {
        const int q   = t + s * 128;
        const int row = q >> 4;         // 0..15
        const int c4  = (q & 15) * 4;   // 0..60
        const size_t g = zoff + (size_t)(yb + row) * XDIM + (xb0 + c4);
        const int l    = row * 76 + 4 + c4;
#ifdef USE_ASYNC_LDS
        __builtin_amdgcn_global_load_async_to_lds_b128(
            (v4i*)(src + g), (lds_v4i*)&smem[l], 0, 0);
#else
        const float4 v      = *(const float4*)(src + g);
        *(float4*)(&smem[l]) = v;
#endif
    }

    // ---- x halo columns (wave 0): threads 0..15 left, 16..31 right ----
    float hval = 0.0f;
    int   hl   = 0;
    if (t < 32) {
        const int row = t & 15;
        const int ghx = (t < 16) ? clampi(xb0 - 1, 0, XDIM - 1)
                                 : clampi(xb0 + 64, 0, XDIM - 1);
        hval = src[zoff + (size_t)(yb + row) * XDIM + ghx];
        hl   = row * 76 + ((t < 16) ? 3 : 68);
    }
#ifdef USE_ASYNC_LDS
    WAIT_ASYNC();
#endif
    if (t < 32) smem[hl] = hval;
    __syncthreads();

    const int xb = xb0 + w * 16;  // wave tile x base
    const int gx = xb + col;

    // ---- y-derivative via V_WMMA_F32_16X16X4_F32, K = 20 (18 halo rows + 2 pad) ----
    // A (16x4 chunk): lane<16 holds row M=col, K = 4c+{0,1}; lane>=16 K = 4c+{2,3}.
    // B (4x16 chunk): same k indexing, value = X[y = yb + k - 1 (clamped), gx].
    v8f acc = {0.f, 0.f, 0.f, 0.f, 0.f, 0.f, 0.f, 0.f};
#pragma unroll
    for (int c = 0; c < 5; ++c) {
        const int k0 = 4 * c + 2 * hi;
        const int k1 = k0 + 1;
        v2f a, b;
        a.x = gcoef(col, k0);
        a.y = gcoef(col, k1);
        const int y0 = clampi(yb + k0 - 1, 0, YDIM - 1);
        const int y1 = clampi(yb + k1 - 1, 0, YDIM - 1);
        b.x = src[zoff + (size_t)y0 * XDIM + gx];
        b.y = src[zoff + (size_t)y1 * XDIM + gx];
        acc = __builtin_amdgcn_wmma_f32_16x16x4_f32(false, a, false, b,
                                                    (short)0, acc, false, false);
    }

    // ---- z-derivative (direct loads), x-derivative (LDS), combine, store ----
    const int   zm = clampi(z - 1, 0, ZDIM - 1);
    const int   zp = clampi(z + 1, 0, ZDIM - 1);
    const float fz = (z == 0 || z == ZDIM - 1) ? 2.0f : 1.0f;
    const float fx = (gx == 0 || gx == XDIM - 1) ? 2.0f : 1.0f;

#pragma unroll
    for (int r = 0; r < 8; ++r) {
        const int ly = r + 8 * hi;  // matches WMMA C/D row layout
        const int gy = yb + ly;
        const size_t rowoff = (size_t)gy * XDIM + gx;

        const float vzm = src[(size_t)zm * plane + rowoff];
        const float vzp = src[(size_t)zp * plane + rowoff];
        const float dzc = 0.5f * (vzp - vzm);

        const int   lb  = ly * 76 + 4 + w * 16 + col;
        const float cv  = smem[lb];
        const float dxc = 0.5f * (smem[lb + 1] - smem[lb - 1]);

        const float fy  = (gy == 0 || gy == YDIM - 1) ? 2.0f : 1.0f;
        const float dyc = acc[r];

        // clamped central diff at a global edge is exactly half the one-sided
        // difference, so the f* = 2 factors reproduce torch.gradient exactly.
        dst[zoff + rowoff] = cv + COEFF * (fx * dxc + fy * dyc + fz * dzc);
    }
}

extern "C" void kernel_launch(void* const* d_in, const int* in_sizes, int n_in,
                              void* d_out, int out_size, void* d_ws, size_t ws_size,
                              hipStream_t stream) {
    const float* in  = (const float*)d_in[0];
    float*       out = (float*)d_out;
    float*       tmp = (float*)d_ws;
    const size_t nbytes = (size_t)ZDIM * YDIM * XDIM * sizeof(float);
    if (ws_size < nbytes) tmp = out;  // should not happen; degenerate fallback

    dim3 grid(XDIM / 64, YDIM / 16, ZDIM);
    dim3 block(128);

    const float* s = in;
    float*       d = tmp;
    for (int it = 0; it < ITERS; ++it) {
        diffusion_step<<<grid, block, 0, stream>>>(s, d);
        s = d;
        d = (d == tmp) ? out : tmp;   // ping-pong; iteration 10 lands in d_out
    }
}